// VoxelOcclusionTester_83013127897913
// MI455X (gfx1250) — compile-verified
//
#include <hip/hip_runtime.h>

typedef __attribute__((ext_vector_type(2))) float v2f;
typedef __attribute__((ext_vector_type(8))) float v8f;

#define IMG_H 512
#define IMG_W 512
#define GRID_HALF 0.025f   // 0.5 * GRID_SIZE
#define OCC_EPS 0.001f

// One wave (32 lanes) handles 16 voxels of one batch (batch = blockIdx.y).
//  - WMMA f32 16x16x4 projects the 16 voxel centers through K:
//      A(16x4): rows 0..2 and 8..10 = rows of K (duplicated so both lane
//               halves receive the projected center in-lane), rest 0.
//      B(4x16): column n = [x_n, y_n, z_n, 0] of voxel n.
//      D(16x16): lane l<16 -> d[0..2] = K.center(voxel l) (rows 0..2);
//                lane l>=16 -> d[0..2] = K.center(voxel l-16) (rows 8..10).
//  - lanes 0..15 test corners 0..3 (dx=-h), lanes 16..31 corners 4..7 (dx=+h),
//    using p(corner) = p(center) + dx*K[:,0] + dy*K[:,1] + dz*K[:,2].
//  - __shfl_xor(16) combines the two 4-corner AND flags; lane<16 stores.
__global__ __launch_bounds__(256)
void voxel_occ_kernel(const float* __restrict__ centers,   // (B, N, 3)
                      const float* __restrict__ mask,      // (B, 512, 512)
                      const float* __restrict__ depth,     // (B, 512, 512)
                      const float* __restrict__ Kmat,      // (3, 3)
                      float* __restrict__ out,             // (B, N)
                      int n_per_b)
{
  const int lane    = threadIdx.x & 31;
  const int wave    = blockIdx.x * (blockDim.x >> 5) + (threadIdx.x >> 5);
  const int halfsel = lane >> 4;        // 0: lanes 0-15, 1: lanes 16-31
  const int col     = lane & 15;        // voxel slot within the wave
  const int b       = blockIdx.y;       // batch index (no per-thread division)

  const int lv = wave * 16 + col;       // voxel id within batch (unclamped)
  int vloc = lv;
  if (vloc >= n_per_b) vloc = n_per_b - 1;  // clamp: duplicate work, store masked
  const int vidx = b * n_per_b + vloc;      // global voxel row

  // ---- B matrix (4x16 f32): column = voxel center, padded with 0 ----
  // VGPR0: lanes0-15 row0 (x), lanes16-31 row2 (z)
  // VGPR1: lanes0-15 row1 (y), lanes16-31 row3 (0)
  const float* cp = centers + (size_t)vidx * 3;
  v2f bmat;
  bmat.x = cp[halfsel ? 2 : 0];
  bmat.y = halfsel ? 0.0f : cp[1];

  // ---- A matrix (16x4 f32): rows {0,1,2} and {8,9,10} = rows of K ----
  // lane l holds row m = l&15: lanes0-15 -> A[m][0],A[m][1];
  //                            lanes16-31 -> A[m][2],A[m][3]
  const int r  = col & 7;               // K row if < 3
  const int ri = r < 3 ? r : 2;         // clamped index (avoid OOB speculation)
  v2f amat;
  amat.x = (r < 3) ? Kmat[ri * 3 + (halfsel ? 2 : 0)] : 0.0f;
  amat.y = (r < 3 && !halfsel) ? Kmat[ri * 3 + 1] : 0.0f;

  v8f acc = {};
  acc = __builtin_amdgcn_wmma_f32_16x16x4_f32(
      /*neg_a=*/false, amat, /*neg_b=*/false, bmat,
      /*c_mod=*/(short)0, acc, /*reuse_a=*/false, /*reuse_b=*/false);

  const float px = acc[0];   // K row0 . center
  const float py = acc[1];   // K row1 . center
  const float pz = acc[2];   // K row2 . center

  // K columns scaled by half-grid (uniform -> scalar-cached loads)
  const float h = GRID_HALF;
  const float cx0 = h * Kmat[0], cx1 = h * Kmat[3], cx2 = h * Kmat[6]; // K[:,0]*h
  const float cy0 = h * Kmat[1], cy1 = h * Kmat[4], cy2 = h * Kmat[7]; // K[:,1]*h
  const float cz0 = h * Kmat[2], cz1 = h * Kmat[5], cz2 = h * Kmat[8]; // K[:,2]*h

  // dx sign: corners 0..3 (lanes<16) have dx=-h; corners 4..7 have dx=+h
  const float sx = halfsel ? 1.0f : -1.0f;
  const float qx = px + sx * cx0;
  const float qy = py + sx * cx1;
  const float qz = pz + sx * cx2;

  const float* __restrict__ mb = mask  + (size_t)b * (IMG_H * IMG_W);
  const float* __restrict__ db = depth + (size_t)b * (IMG_H * IMG_W);

  bool occ = true;
#pragma unroll
  for (int j = 0; j < 4; ++j) {          // corner c = halfsel*4 + j
    const float sy = (j & 2) ? 1.0f : -1.0f;   // dy = go[(c>>1)&1]
    const float sz = (j & 1) ? 1.0f : -1.0f;   // dz = go[c&1]
    const float fx = qx + sy * cy0 + sz * cz0;
    const float fy = qy + sy * cy1 + sz * cz1;
    const float fz = qz + sy * cy2 + sz * cz2;
    const float rz = __builtin_amdgcn_rcpf(fz);   // v_rcp_f32 + mul, not full div
    const float fu = fx * rz;
    const float fv = fy * rz;
    const bool in_img = (fu >= 0.0f) & (fu <= (float)(IMG_W - 1)) &
                        (fv >= 0.0f) & (fv <= (float)(IMG_H - 1)) &
                        (fz > 0.0f);
    int ui = (int)rintf(fu);             // round-half-even == jnp.round
    int vi = (int)rintf(fv);
    ui = min(max(ui, 0), IMG_W - 1);
    vi = min(max(vi, 0), IMG_H - 1);
    const int pix = vi * IMG_W + ui;
    const float mval = mb[pix];
    const float dval = db[pix];
    occ = occ & in_img & (mval > 0.5f) & (dval < fz - OCC_EPS);
  }

  // AND the two half-wave flags (corners 0-3 with corners 4-7)
  const unsigned occ_u = occ ? 1u : 0u;
  const unsigned other = __shfl_xor(occ_u, 16, 32);
  if (halfsel == 0 && lv < n_per_b) {
    out[(size_t)b * n_per_b + lv] = (occ_u & other) ? 1.0f : 0.0f;
  }
}

extern "C" void kernel_launch(void* const* d_in, const int* in_sizes, int n_in,
                              void* d_out, int out_size, void* d_ws, size_t ws_size,
                              hipStream_t stream) {
  const float* centers = (const float*)d_in[0];  // (B, N, 3) f32
  const float* mask    = (const float*)d_in[1];  // (B, 512, 512) f32
  const float* depth   = (const float*)d_in[2];  // (B, 512, 512) f32
  const float* Kmat    = (const float*)d_in[3];  // (3, 3) f32
  float* out = (float*)d_out;                    // (B, N) f32

  const int n_total = in_sizes[0] / 3;                       // B*N
  const int B       = in_sizes[1] / (IMG_H * IMG_W);
  const int n_per_b = n_total / B;                           // N

  const int waves_per_batch  = (n_per_b + 15) / 16;
  const int threads_per_batch = waves_per_batch * 32;
  const int block = 256;
  const int blocks_per_batch = (threads_per_batch + block - 1) / block;

  dim3 grid(blocks_per_batch, B, 1);
  voxel_occ_kernel<<<grid, block, 0, stream>>>(centers, mask, depth, Kmat,
                                               out, n_per_b);
}